// DiscoModel_26242250179146
// MI455X (gfx1250) — compile-verified
//
#include <hip/hip_runtime.h>
#include <math.h>

typedef __attribute__((ext_vector_type(2))) float v2f;
typedef __attribute__((ext_vector_type(8))) float v8f;

#define EPS 1e-12f
#define CHUNK 2048   // columns staged in LDS per tile

// ---- WMMA row-sum: D = A(16x4) * ones(4x16) + C  -> every column of D = row sum
__device__ __forceinline__ v8f wmma_rowsum(v2f a, v8f c) {
    v2f ones = {1.0f, 1.0f};
    return __builtin_amdgcn_wmma_f32_16x16x4_f32(
        /*neg_a=*/false, a, /*neg_b=*/false, ones,
        /*c_mod=*/(short)0, c, /*reuse_a=*/false, /*reuse_b=*/false);
}

__device__ __forceinline__ v2f vabs2(v2f v) {
    v2f r; r.x = __builtin_fabsf(v.x); r.y = __builtin_fabsf(v.y); return r;
}

// ---------------------------------------------------------------------------
// Kernel 1: amatavg[i] = (1/N) sum_j |x_i - x_j| w_j   (and same for y)
// 128 threads = 4 waves; each wave owns 16 rows; WMMA accumulates row sums.
// ---------------------------------------------------------------------------
__global__ __launch_bounds__(128) void k_rowmeans(
    const float* __restrict__ x, const float* __restrict__ y,
    const float* __restrict__ w, float* __restrict__ ax, float* __restrict__ bx,
    int n)
{
    __shared__ __align__(16) float s_x[CHUNK];
    __shared__ __align__(16) float s_y[CHUNK];
    __shared__ __align__(16) float s_w[CHUNK];

    const int lane    = threadIdx.x & 31;
    const int wv      = threadIdx.x >> 5;
    const int rowbase = blockIdx.x * 64 + wv * 16;
    const int m       = lane & 15;
    const int klo     = (lane & 16) ? 2 : 0;   // ISA 16x4 f32 A-layout

    const float xs = x[rowbase + m];
    const float ys = y[rowbase + m];
    const v2f  xi = {xs, xs};
    const v2f  yi = {ys, ys};

    v8f accX = {};
    v8f accY = {};

    for (int j0 = 0; j0 < n; j0 += CHUNK) {
        int len = n - j0; if (len > CHUNK) len = CHUNK;
        __syncthreads();
        for (int t = threadIdx.x; t < len; t += 128) {
            s_x[t] = x[j0 + t];
            s_y[t] = y[j0 + t];
            s_w[t] = w[j0 + t];
        }
        __syncthreads();
#pragma unroll 4
        for (int jj = 0; jj < len; jj += 4) {
            const int jA = jj + klo;               // even -> 8B aligned
            v2f xj = *(const v2f*)&s_x[jA];
            v2f yj = *(const v2f*)&s_y[jA];
            v2f wj = *(const v2f*)&s_w[jA];
            v2f fx = vabs2(xi - xj) * wj;
            v2f fy = vabs2(yi - yj) * wj;
            accX = wmma_rowsum(fx, accX);
            accY = wmma_rowsum(fy, accY);
        }
    }

    const float invN = 1.0f / (float)n;
    if (lane == 0 || lane == 16) {
        const int rb = rowbase + ((lane == 16) ? 8 : 0);
#pragma unroll
        for (int r = 0; r < 8; ++r) {
            ax[rb + r] = accX[r] * invN;
            bx[rb + r] = accY[r] * invN;
        }
    }
}

// ---------------------------------------------------------------------------
// Kernel 2: grand means  am = mean(amatavg*w), bm = mean(bmatavg*w)
// ---------------------------------------------------------------------------
__global__ __launch_bounds__(256) void k_gmean(
    const float* __restrict__ ax, const float* __restrict__ bx,
    const float* __restrict__ w, float* __restrict__ scal, int n)
{
    __shared__ float sa[256];
    __shared__ float sb[256];
    const int tid = threadIdx.x;
    float pa = 0.0f, pb = 0.0f;
    for (int i = tid; i < n; i += 256) {
        float wi = w[i];
        pa += ax[i] * wi;
        pb += bx[i] * wi;
    }
    sa[tid] = pa; sb[tid] = pb;
    __syncthreads();
    for (int s = 128; s > 0; s >>= 1) {
        if (tid < s) { sa[tid] += sa[tid + s]; sb[tid] += sb[tid + s]; }
        __syncthreads();
    }
    if (tid == 0) {
        scal[0] = sa[0] / (float)n;   // am
        scal[1] = sb[0] / (float)n;   // bm
    }
}

// ---------------------------------------------------------------------------
// Kernel 3: per-row raw sums of centered products
//   A_ij = |x_i-x_j| - ax_j + (am - ax_i),  B_ij analogous
//   sab[i] = sum_j A*B*w,  saa[i] = sum_j A*A*w,  sbb[i] = sum_j B*B*w
// ---------------------------------------------------------------------------
__global__ __launch_bounds__(128) void k_center(
    const float* __restrict__ x,  const float* __restrict__ y,
    const float* __restrict__ w,  const float* __restrict__ ax,
    const float* __restrict__ bx, const float* __restrict__ scal,
    float* __restrict__ sab, float* __restrict__ saa, float* __restrict__ sbb,
    int n)
{
    __shared__ __align__(16) float s_x[CHUNK];
    __shared__ __align__(16) float s_y[CHUNK];
    __shared__ __align__(16) float s_w[CHUNK];
    __shared__ __align__(16) float s_ax[CHUNK];
    __shared__ __align__(16) float s_bx[CHUNK];

    const int lane    = threadIdx.x & 31;
    const int wv      = threadIdx.x >> 5;
    const int rowbase = blockIdx.x * 64 + wv * 16;
    const int m       = lane & 15;
    const int klo     = (lane & 16) ? 2 : 0;

    const float am = scal[0];
    const float bm = scal[1];
    const int   row = rowbase + m;
    const float xs = x[row], ys = y[row];
    const float cAs = am - ax[row];
    const float cBs = bm - bx[row];
    const v2f xi = {xs, xs}, yi = {ys, ys};
    const v2f cA = {cAs, cAs}, cB = {cBs, cBs};

    v8f accAB = {};
    v8f accAA = {};
    v8f accBB = {};

    for (int j0 = 0; j0 < n; j0 += CHUNK) {
        int len = n - j0; if (len > CHUNK) len = CHUNK;
        __syncthreads();
        for (int t = threadIdx.x; t < len; t += 128) {
            s_x[t]  = x[j0 + t];
            s_y[t]  = y[j0 + t];
            s_w[t]  = w[j0 + t];
            s_ax[t] = ax[j0 + t];
            s_bx[t] = bx[j0 + t];
        }
        __syncthreads();
#pragma unroll 4
        for (int jj = 0; jj < len; jj += 4) {
            const int jA = jj + klo;
            v2f xj  = *(const v2f*)&s_x[jA];
            v2f yj  = *(const v2f*)&s_y[jA];
            v2f wj  = *(const v2f*)&s_w[jA];
            v2f axj = *(const v2f*)&s_ax[jA];
            v2f bxj = *(const v2f*)&s_bx[jA];

            v2f aA = vabs2(xi - xj) - axj + cA;
            v2f aB = vabs2(yi - yj) - bxj + cB;
            v2f aw = aA * wj;

            accAB = wmma_rowsum(aw * aB, accAB);
            accAA = wmma_rowsum(aw * aA, accAA);
            accBB = wmma_rowsum(aB * aB * wj, accBB);
        }
    }

    if (lane == 0 || lane == 16) {
        const int rb = rowbase + ((lane == 16) ? 8 : 0);
#pragma unroll
        for (int r = 0; r < 8; ++r) {
            sab[rb + r] = accAB[r];
            saa[rb + r] = accAA[r];
            sbb[rb + r] = accBB[r];
        }
    }
}

// ---------------------------------------------------------------------------
// Kernel 4: final scalar
// ---------------------------------------------------------------------------
__global__ __launch_bounds__(256) void k_final(
    const float* __restrict__ sab, const float* __restrict__ saa,
    const float* __restrict__ sbb, const float* __restrict__ w,
    const int* __restrict__ powp, float* __restrict__ out, int n)
{
    __shared__ float s0[256];
    __shared__ float s1[256];
    __shared__ float s2[256];
    const int tid = threadIdx.x;
    float p0 = 0.0f, p1 = 0.0f, p2 = 0.0f;
    for (int i = tid; i < n; i += 256) {
        float wi = w[i];
        p0 += __builtin_fabsf(sab[i]) * wi;
        p1 += saa[i] * wi;
        p2 += sbb[i] * wi;
    }
    s0[tid] = p0; s1[tid] = p1; s2[tid] = p2;
    __syncthreads();
    for (int s = 128; s > 0; s >>= 1) {
        if (tid < s) { s0[tid] += s0[tid+s]; s1[tid] += s1[tid+s]; s2[tid] += s2[tid+s]; }
        __syncthreads();
    }
    if (tid == 0) {
        const float n2  = (float)n * (float)n;
        const float num = s0[0] / n2;
        const float mAA = s1[0] / n2;
        const float mBB = s2[0] / n2;
        const float den = __builtin_fabsf(mAA * mBB);
        const int   p   = powp[0];
        float d;
        if (p == 1)      d = num / sqrtf(den + EPS);
        else if (p == 2) d = (num * num) / (den + EPS);
        else             d = powf(num / sqrtf(mAA * mBB) + EPS, (float)p);
        if (isnan(d)) d = 0.0f;
        d = fmaxf(d, 0.0f);
        out[0] = d;
    }
}

// ---------------------------------------------------------------------------
extern "C" void kernel_launch(void* const* d_in, const int* in_sizes, int n_in,
                              void* d_out, int out_size, void* d_ws, size_t ws_size,
                              hipStream_t stream) {
    const float* x = (const float*)d_in[0];   // var_1
    const float* y = (const float*)d_in[1];   // var_2
    const float* w = (const float*)d_in[2];   // normedweight
    const int*   p = (const int*)d_in[3];     // power
    float* out = (float*)d_out;
    const int n = in_sizes[0];                // 8192

    float* ws   = (float*)d_ws;               // needs (5n+2)*4 bytes (~160KB)
    float* ax   = ws;
    float* bx   = ws + 1 * n;
    float* sab  = ws + 2 * n;
    float* saa  = ws + 3 * n;
    float* sbb  = ws + 4 * n;
    float* scal = ws + 5 * n;

    const int grid = (n + 63) / 64;           // 64 rows per 128-thread block

    k_rowmeans<<<grid, 128, 0, stream>>>(x, y, w, ax, bx, n);
    k_gmean  <<<1, 256, 0, stream>>>(ax, bx, w, scal, n);
    k_center <<<grid, 128, 0, stream>>>(x, y, w, ax, bx, scal, sab, saa, sbb, n);
    k_final  <<<1, 256, 0, stream>>>(sab, saa, sbb, w, p, out, n);
}